// MARMeansCovsLayer_24541443129968
// MI455X (gfx1250) — compile-verified
//
#include <hip/hip_runtime.h>

typedef __attribute__((ext_vector_type(2))) float v2f;
typedef __attribute__((ext_vector_type(8))) float v8f;

#define T_FULL 8192
#define CDIM   16
#define LLAGS  5
#define NB     4
#define TOUT   8186   // T - n_lags - 1

// One wave32 per output timestep. mu[b,t] computed with 20 accumulating
// v_wmma_f32_16x16x4_f32 ops (5 lags x K=16 in chunks of 4), exact f32 math.
__global__ __launch_bounds__(256)
void mar_mu_wmma(const float* __restrict__ data,
                 const float* __restrict__ coeffs,
                 float* __restrict__ mu_out)
{
    const int lane = threadIdx.x & 31;
    const int wave = threadIdx.x >> 5;
    const int wid  = blockIdx.x * 8 + wave;     // global wave id = output row id
    const int total = NB * TOUT;
    if (wid >= total) return;                   // uniform per wave: EXEC stays all-1s

    const int b = wid / TOUT;
    const int k = wid - b * TOUT;
    const int t = k + LLAGS;                    // 5 .. T-2

    const int row = lane & 15;                  // A-matrix row M
    const int hi  = lane >> 4;                  // 0 -> K={0,1}, 1 -> K={2,3}

    v8f acc = {};
    #pragma unroll
    for (int lag = 0; lag < LLAGS; ++lag) {
        int tc = t + (lag * (lag - 1)) / 2;     // coeff time offsets {0,0,1,3,6}
        if (tc >= T_FULL) tc -= T_FULL;         // wraps for t near T-2 (roll semantics)
        const int td = t - lag;                 // data time, never wraps (t >= 5)

        const float* cb = coeffs + ((((size_t)b * T_FULL + tc) * LLAGS + lag) << 8);
        const float* db = data   + (((size_t)b * T_FULL + td) << 4);

        #pragma unroll
        for (int c = 0; c < 4; ++c) {
            // A[M=row][K=4c + 2*hi + {0,1}]  : each matrix element loaded exactly once
            v2f a  = *(const v2f*)(cb + row * CDIM + c * 4 + hi * 2);
            // B[K][N] = d[K] for every column N (vector replicated across columns)
            v2f bv = *(const v2f*)(db + c * 4 + hi * 2);
            acc = __builtin_amdgcn_wmma_f32_16x16x4_f32(
                    /*neg_a=*/false, a, /*neg_b=*/false, bv,
                    /*c_mod=*/(short)0, acc, /*reuse_a=*/false, /*reuse_b=*/false);
        }
    }

    // Every column of D equals mu: lane 0 VGPRs 0..7 = mu[0..7],
    // lane 16 VGPRs 0..7 = mu[8..15].
    if (row == 0) {
        float* p = mu_out + ((size_t)wid << 4) + hi * 8;
        *(float4*)(p)     = make_float4(acc[0], acc[1], acc[2], acc[3]);
        *(float4*)(p + 4) = make_float4(acc[4], acc[5], acc[6], acc[7]);
    }
}

// Contiguous per-batch segment copy, float4 (b128) streams.
// dst[b*dstStride4 + i] = src[b*srcStride4 + i], i < n4, b = blockIdx.y
__global__ __launch_bounds__(256)
void batched_copy_f4(float4* __restrict__ dst, const float4* __restrict__ src,
                     long n4, long dstStride4, long srcStride4)
{
    long i = (long)blockIdx.x * blockDim.x + threadIdx.x;
    int  b = blockIdx.y;
    if (i < n4)
        dst[(size_t)b * dstStride4 + i] = src[(size_t)b * srcStride4 + i];
}

extern "C" void kernel_launch(void* const* d_in, const int* in_sizes, int n_in,
                              void* d_out, int out_size, void* d_ws, size_t ws_size,
                              hipStream_t stream) {
    const float* data   = (const float*)d_in[0];   // (4, 8192, 16)
    const float* coeffs = (const float*)d_in[1];   // (4, 8192, 5, 16, 16)
    const float* covs   = (const float*)d_in[2];   // (4, 8192, 16, 16)
    // d_in[3] = n_lags scalar (5), baked into constants.

    float* out    = (float*)d_out;
    float* x_out  = out;                                   // (4, 8186, 16)
    float* mu_out = out + (size_t)NB * TOUT * CDIM;        // (4, 8186, 16)
    float* sg_out = out + 2 * (size_t)NB * TOUT * CDIM;    // (4, 8186, 16, 16)

    // --- mu via WMMA: 4*8186 waves, 8 waves per 256-thread block (divides exactly)
    {
        dim3 blk(256);
        dim3 grd((NB * TOUT + 7) / 8);
        mar_mu_wmma<<<grd, blk, 0, stream>>>(data, coeffs, mu_out);
    }

    // --- x_t = data[:, 6:8192, :]  (contiguous per batch, 16B-aligned offsets)
    {
        long n4 = (long)TOUT * CDIM / 4;                   // 32744 float4 per batch
        dim3 g((unsigned)((n4 + 255) / 256), NB);
        batched_copy_f4<<<g, 256, 0, stream>>>(
            (float4*)x_out, (const float4*)(data + 6 * CDIM),
            n4, n4, (long)T_FULL * CDIM / 4);
    }

    // --- sigma_t = covs[:, 5:8191, :, :]  (contiguous per batch)
    {
        long n4 = (long)TOUT * CDIM * CDIM / 4;            // 523904 float4 per batch
        dim3 g((unsigned)((n4 + 255) / 256), NB);
        batched_copy_f4<<<g, 256, 0, stream>>>(
            (float4*)sg_out, (const float4*)(covs + 5 * CDIM * CDIM),
            n4, n4, (long)T_FULL * CDIM * CDIM / 4);
    }
}